// NormPolyKernelTorch_2783138808483
// MI455X (gfx1250) — compile-verified
//
#include <hip/hip_runtime.h>

// Problem constants (fixed by the reference): X,Y are 128 x 8192 f32.
#define NFEAT 128
#define NCOLS 8192
#define C2    100.0f   // C*C with C = 10.0

typedef __attribute__((ext_vector_type(2))) float v2f;
typedef __attribute__((ext_vector_type(8))) float v8f;

// ---------------------------------------------------------------------------
// Kernel 1: per-column scale  s[n] = 1/sqrt((||col_n||^2 + c2)^3)
// One thread per column; loads are coalesced across threads (n is fast dim).
// ---------------------------------------------------------------------------
__global__ __launch_bounds__(256)
void colscale_kernel(const float* __restrict__ M, float* __restrict__ s) {
    const int n = blockIdx.x * blockDim.x + threadIdx.x;
    float acc = 0.0f;
#pragma unroll 8
    for (int k = 0; k < NFEAT; ++k) {
        const float v = M[(size_t)k * NCOLS + n];
        acc = fmaf(v, v, acc);
    }
    const float t = acc + C2;
    s[n] = rsqrtf(t * t * t);
}

// ---------------------------------------------------------------------------
// Kernel 2: f32 WMMA GEMM  K[n,m] = sum_k X[k,n]*Y[k,m], fused epilogue
//           out[n,m] = s1[n] * (K + c2)^3 * s2[m]
//
// Block = 256 threads = 8 waves, computes a 128x128 output tile.
// Wave w computes rows [wtile*16, wtile*16+16) x 128 cols = 8 WMMA tiles.
// A-frag (16x4 f32, ISA 7.12.2): lane<16 -> M=lane, K={0,1}; lane>=16 -> K={2,3}.
// B-frag (4x16 f32) mirrored:    lane<16 -> N=lane, K={0,1}; lane>=16 -> K={2,3}.
// C/D (16x16 f32): VGPR r, lane L -> row r + (L>=16 ? 8 : 0), col L&15.
// ---------------------------------------------------------------------------
__global__ __launch_bounds__(256, 2)
void poly_gemm_kernel(const float* __restrict__ X, const float* __restrict__ Y,
                      const float* __restrict__ s1, const float* __restrict__ s2,
                      float* __restrict__ out) {
    const int lane = threadIdx.x & 31;
    const int wave = threadIdx.x >> 5;
    const int m    = lane & 15;   // row (A) / col (B) within 16-tile
    const int kh   = lane >> 4;   // K half-select: 0 -> K{0,1}, 1 -> K{2,3}

    const int rowA = blockIdx.y * 128 + wave * 16 + m;  // n index for A fragment
    const int colB = blockIdx.x * 128 + m;              // base m index for B fragments

    // Pointers to K-step 0: a.x <- X[2*kh + 0, rowA], a.y <- X[2*kh + 1, rowA]
    const float* xp = X + (size_t)(2 * kh) * NCOLS + rowA;
    const float* yp = Y + (size_t)(2 * kh) * NCOLS + colB;

    v8f acc[8];
#pragma unroll
    for (int j = 0; j < 8; ++j) acc[j] = (v8f){0.f, 0.f, 0.f, 0.f, 0.f, 0.f, 0.f, 0.f};

    // K = 128 -> 32 steps of 16x16x4 WMMA per column tile.
#pragma unroll 2
    for (int kk = 0; kk < NFEAT / 4; ++kk) {
        v2f a;
        a.x = xp[0];
        a.y = xp[NCOLS];
#pragma unroll
        for (int j = 0; j < 8; ++j) {
            v2f b;
            b.x = yp[j * 16];
            b.y = yp[NCOLS + j * 16];
            acc[j] = __builtin_amdgcn_wmma_f32_16x16x4_f32(
                /*neg_a=*/false, a, /*neg_b=*/false, b,
                /*c_mod=*/(short)0, acc[j],
                /*reuse_a=*/false, /*reuse_b=*/false);
        }
        xp += (size_t)4 * NCOLS;
        yp += (size_t)4 * NCOLS;
    }

    // Fused epilogue: out[n,m] = s1[n] * (acc + c2)^3 * s2[m]
    const int rowS = blockIdx.y * 128 + wave * 16 + kh * 8;  // first row this lane stores
    float s1v[8];
#pragma unroll
    for (int r = 0; r < 8; ++r) s1v[r] = s1[rowS + r];

#pragma unroll
    for (int j = 0; j < 8; ++j) {
        const int   col = blockIdx.x * 128 + j * 16 + (lane & 15);
        const float s2v = s2[col];
        float* op = out + (size_t)rowS * NCOLS + col;
#pragma unroll
        for (int r = 0; r < 8; ++r) {
            const float t = acc[j][r] + C2;
            op[(size_t)r * NCOLS] = s1v[r] * (t * t * t) * s2v;
        }
    }
}

// ---------------------------------------------------------------------------
extern "C" void kernel_launch(void* const* d_in, const int* in_sizes, int n_in,
                              void* d_out, int out_size, void* d_ws, size_t ws_size,
                              hipStream_t stream) {
    const float* X = (const float*)d_in[0];
    const float* Y = (const float*)d_in[1];
    float* s1 = (float*)d_ws;          // 8192 floats
    float* s2 = s1 + NCOLS;            // 8192 floats (64 KB total ws use)

    colscale_kernel<<<NCOLS / 256, 256, 0, stream>>>(X, s1);
    colscale_kernel<<<NCOLS / 256, 256, 0, stream>>>(Y, s2);

    dim3 grid(NCOLS / 128, NCOLS / 128);   // 64 x 64 blocks of 128x128 tiles
    poly_gemm_kernel<<<grid, 256, 0, stream>>>(X, Y, s1, s2, (float*)d_out);
}